// BatchedDistanceGradientNet_876173328998
// MI455X (gfx1250) — compile-verified
//
#include <hip/hip_runtime.h>
#include <hip/hip_bf16.h>

typedef __attribute__((ext_vector_type(2))) float v2f;
typedef __attribute__((ext_vector_type(8))) float v8f;

#define NP 4096           // points
#define NO 32             // zonotopes
#define NH 26             // hyperplanes per zonotope
#define NV 48             // edges per zonotope
#define NFLAT (NO * NH)   // 832 flattened hyperplanes (= 52 * 16 exactly)
#define NTILES (NFLAT / 16)
#define PTS_PER_BLOCK 16
#define THREADS 512       // 16 wave32 waves
#define EPSV 1e-4f

__global__ __launch_bounds__(THREADS)
void zono_dist_kernel(const float* __restrict__ point,
                      const float* __restrict__ hA,   // [O][H][3]
                      const float* __restrict__ hb,   // [O][H]
                      const float* __restrict__ v1p,  // [O][V][3]
                      const float* __restrict__ v2p,  // [O][V][3]
                      float* __restrict__ out_dist,   // [P][O]
                      float* __restrict__ out_grad)   // [P][O][3]
{
    // a[p_local][g] = point[p]·A[g] - b[g], produced by WMMA, consumed in phase 2
    __shared__ float sApb[PTS_PER_BLOCK][NFLAT];      // 16 x 832 f32 = 53 KB

    const int tid  = threadIdx.x;
    const int lane = tid & 31;
    const int wave = tid >> 5;
    const int m    = lane & 15;
    const bool hi  = (lane >= 16);
    const int p0   = blockIdx.x * PTS_PER_BLOCK;
    const float INF = __builtin_inff();

    // ---------------- Phase 1: V_WMMA_F32_16X16X4_F32 -----------------------
    // A-matrix 16x4 (M=16 points, K=[x,y,z,1]):
    //   lanes 0-15 : VGPR0=K0(x), VGPR1=K1(y);  lanes 16-31: VGPR0=K2(z), VGPR1=K3(1)
    // Branchless: always load, select addresses/values with cndmask (no exec
    // divergence inside the WMMA loop).
    const float* pbase = point + (p0 + m) * 3;
    v2f afrag;
    {
        const float ax = pbase[hi ? 2 : 0];  // K0 (x) or K2 (z)
        const float ay = pbase[1];           // K1 (y), dead for hi lanes
        afrag.x = ax;
        afrag.y = hi ? 1.0f : ay;            // K3 bias lane multiplies -b row
    }

    for (int t = wave; t < NTILES; t += (THREADS / 32)) {
        const int g = t * 16 + m;            // flattened hyperplane index < 832
        // B-matrix 4x16 (K rows = [Ax,Ay,Az,-b], N=16 hyperplanes), mirrored layout
        const float* gbase = hA + g * 3;
        const float b0 = gbase[hi ? 2 : 0];  // K0 (Ax) or K2 (Az)
        const float t1 = gbase[1];           // K1 (Ay), dead for hi lanes
        const float t2 = hb[g];              // bias,     dead for lo lanes
        v2f bfrag;
        bfrag.x = b0;
        bfrag.y = hi ? -t2 : t1;             // K3 row = -b, K1 row = Ay
        v8f c = {};
        c = __builtin_amdgcn_wmma_f32_16x16x4_f32(false, afrag, false, bfrag,
                                                  (short)0, c, false, false);
        // D layout: VGPR r -> M = r (lanes 0-15) or r+8 (lanes 16-31), N = lane%16
        const int rbase = hi ? 8 : 0;
        #pragma unroll
        for (int r = 0; r < 8; ++r)
            sApb[rbase + r][t * 16 + m] = c[r];
    }
    __syncthreads();

    // ---------------- Phase 2: per-(point, zonotope) reductions -------------
    const int pl = wave;                     // point within block
    const int o  = lane;                     // zonotope
    const int p  = p0 + pl;
    const float px = point[p * 3 + 0];
    const float py = point[p * 3 + 1];
    const float pz = point[p * 3 + 2];

    // a[h] with nan_to_num(nan -> -inf, +inf -> -inf)
    float a[NH];
    bool  isneg = true;
    float maxv  = -INF;
    int   maxi  = 0;
    #pragma unroll
    for (int h = 0; h < NH; ++h) {
        float ah = sApb[pl][o * NH + h];
        if ((ah != ah) || (ah == INF)) ah = -INF;
        a[h] = ah;
        if (ah > 0.0f) isneg = false;
        if (ah > maxv) { maxv = ah; maxi = h; }  // strict > keeps first occurrence
    }

    // face distances: project onto each hyperplane, test vs all hyperplanes
    float minv = INF;
    int   mini = 0;
    #pragma unroll
    for (int h = 0; h < NH; ++h) {
        const float ah  = a[h];
        const float Ahx = hA[(o * NH + h) * 3 + 0];
        const float Ahy = hA[(o * NH + h) * 3 + 1];
        const float Ahz = hA[(o * NH + h) * 3 + 2];
        const float ppx = px - ah * Ahx;
        const float ppy = py - ah * Ahy;
        const float ppz = pz - ah * Ahz;
        bool onz = true;
        for (int j = 0; j < NH; ++j) {
            const float vj = ppx * hA[(o * NH + j) * 3 + 0]
                           + ppy * hA[(o * NH + j) * 3 + 1]
                           + ppz * hA[(o * NH + j) * 3 + 2]
                           - hb[o * NH + j];
            // nan_to_num(nan,+inf -> -inf) then (<= EPS)
            const bool ok = (vj <= EPSV) || (vj != vj) || (vj == INF);
            onz = onz && ok;
        }
        const float dx = ah * Ahx, dy = ah * Ahy, dz = ah * Ahz;
        float perp = sqrtf(dx * dx + dy * dy + dz * dz);
        if (!onz || (perp != perp)) perp = INF;
        if (perp < minv) { minv = perp; mini = h; }
    }

    // edge distances over 48 segments
    float ev = INF;
    float vsx = 0.0f, vsy = 0.0f, vsz = 0.0f;
    for (int e = 0; e < NV; ++e) {
        const int base = (o * NV + e) * 3;
        const float v1x = v1p[base + 0], v1y = v1p[base + 1], v1z = v1p[base + 2];
        const float ex = v2p[base + 0] - v1x;
        const float ey = v2p[base + 1] - v1y;
        const float ez = v2p[base + 2] - v1z;
        const float denom = ex * ex + ey * ey + ez * ez;
        const float th = ((px - v1x) * ex + (py - v1y) * ey + (pz - v1z) * ez) / denom;
        const float ts = fminf(fmaxf(th, 0.0f), 1.0f);
        const float vx = v1x + ts * ex;
        const float vy = v1y + ts * ey;
        const float vz = v1z + ts * ez;
        const float dx = px - vx, dy = py - vy, dz = pz - vz;
        float ed = sqrtf(dx * dx + dy * dy + dz * dz);
        if (ed != ed) ed = INF;
        if (ed < ev) { ev = ed; vsx = vx; vsy = vy; vsz = vz; }
    }

    // final select between face result and edge result
    float dist = isneg ? maxv : minv;
    const int fidx = isneg ? maxi : mini;
    float gx = hA[(o * NH + fidx) * 3 + 0];
    float gy = hA[(o * NH + fidx) * 3 + 1];
    float gz = hA[(o * NH + fidx) * 3 + 2];
    const bool use_edge = (ev < dist) && !isneg;
    if (use_edge) {
        dist = ev;
        gx = (px - vsx) / ev;
        gy = (py - vsy) / ev;
        gz = (pz - vsz) / ev;
    }

    out_dist[p * NO + o] = dist;
    float* g = out_grad + (size_t)(p * NO + o) * 3;
    g[0] = gx;
    g[1] = gy;
    g[2] = gz;
}

extern "C" void kernel_launch(void* const* d_in, const int* in_sizes, int n_in,
                              void* d_out, int out_size, void* d_ws, size_t ws_size,
                              hipStream_t stream) {
    const float* point = (const float*)d_in[0];
    const float* hA    = (const float*)d_in[1];
    const float* hb    = (const float*)d_in[2];
    const float* v1    = (const float*)d_in[3];
    const float* v2    = (const float*)d_in[4];

    float* out_dist = (float*)d_out;              // P*O distances
    float* out_grad = out_dist + (size_t)NP * NO; // P*O*3 gradients

    dim3 grid(NP / PTS_PER_BLOCK);                // 256 blocks
    dim3 block(THREADS);                          // 512 threads = 16 waves
    hipLaunchKernelGGL(zono_dist_kernel, grid, block, 0, stream,
                       point, hA, hb, v1, v2, out_dist, out_grad);
}